// GraphAttention_38774964748855
// MI455X (gfx1250) — compile-verified
//
#include <hip/hip_runtime.h>
#include <math.h>

typedef float v2f __attribute__((ext_vector_type(2)));
typedef float v8f __attribute__((ext_vector_type(8)));
typedef unsigned int v4u __attribute__((ext_vector_type(4)));
typedef int v4i __attribute__((ext_vector_type(4)));
typedef int v8i __attribute__((ext_vector_type(8)));

#define NN 100000
#define EE 800000
#define DD 128

#if defined(__has_builtin)
#if __has_builtin(__builtin_amdgcn_tensor_load_to_lds)
#define HAVE_TDM 1
#endif
#endif

__device__ __forceinline__ unsigned fenc(float x) {
    unsigned b = __float_as_uint(x);
    return (b & 0x80000000u) ? ~b : (b | 0x80000000u);
}
__device__ __forceinline__ float fdec(unsigned u) {
    return (u & 0x80000000u) ? __uint_as_float(u ^ 0x80000000u) : __uint_as_float(~u);
}

#ifdef HAVE_TDM
// Build a Tensor-DMA descriptor (CDNA5 ISA) for a 2D tile of
// f32 rows and issue TENSOR_LOAD_TO_LDS. Must be executed by exactly one wave.
__device__ __forceinline__ void tdm_load_rows(unsigned lds_off, const void* gptr,
                                              unsigned rows, unsigned row_elems,
                                              unsigned stride_elems) {
    unsigned long long ga = (unsigned long long)gptr;
    v4u g0;
    g0.x = 1u;                                              // count=1 valid descriptor
    g0.y = lds_off;                                         // lds_addr (bytes)
    g0.z = (unsigned)ga;                                    // global_addr[31:0]
    g0.w = (unsigned)((ga >> 32) & 0x01FFFFFFu) | (2u << 30); // addr[56:32] | type=2

    v8i g1;
    g1[0] = (int)(2u << 16);                                // data_size = 4 bytes
    g1[1] = (int)((row_elems & 0xFFFFu) << 16);             // tensor_dim0[15:0]
    g1[2] = (int)(((row_elems >> 16) & 0xFFFFu) |           // tensor_dim0[31:16]
                  ((rows & 0xFFFFu) << 16));                // tensor_dim1[15:0]
    g1[3] = (int)(((rows >> 16) & 0xFFFFu) |                // tensor_dim1[31:16]
                  ((row_elems & 0xFFFFu) << 16));           // tile_dim0
    g1[4] = (int)(rows & 0xFFFFu);                          // tile_dim1 (tile_dim2=0)
    g1[5] = (int)stride_elems;                              // tensor_dim0_stride[31:0]
    g1[6] = 0;
    g1[7] = 0;

    v4i z4 = {0, 0, 0, 0};
#if __has_include(<hip/amd_detail/amd_gfx1250_TDM.h>)
    v8i z8 = {0, 0, 0, 0, 0, 0, 0, 0};
    __builtin_amdgcn_tensor_load_to_lds(g0, g1, z4, z4, z8, 0);   // clang-23 6-arg form
#else
    __builtin_amdgcn_tensor_load_to_lds(g0, g1, z4, z4, 0);       // ROCm 7.2 5-arg form
#endif
}
#endif

// Stage `rows` x `row_elems` f32 tile (contiguous rows, leading dim = stride)
// from global into LDS, visible to the whole block afterwards.
__device__ __forceinline__ void stage_rows(float* s, const float* g,
                                           unsigned rows, unsigned row_elems,
                                           unsigned stride, int tid, int nthreads) {
#ifdef HAVE_TDM
    if (tid < 32) {   // wave 0 issues the DMA (TDM ignores EXEC; one issue per wave)
        tdm_load_rows((unsigned)(uintptr_t)s, g, rows, row_elems, stride);
        __builtin_amdgcn_s_wait_tensorcnt(0);
    }
#else
    unsigned n = rows * row_elems;
    for (unsigned i = tid; i < n; i += nthreads)
        s[i] = g[(i / row_elems) * stride + (i % row_elems)];
#endif
    __syncthreads();
}

// One wave computes a 16x16 f32 tile: C += A(16xK) * B(Kx16), via V_WMMA_F32_16X16X4_F32.
// A row-major with leading dim lda (LDS or global), B row-major with leading dim ldb.
// Per ISA 7.12.2: A 16x4 frag: lanes 0-15 hold K={0,1}, lanes 16-31 hold K={2,3}.
__device__ __forceinline__ v8f gemm16x16(const float* __restrict__ A, int lda,
                                         const float* __restrict__ B, int ldb,
                                         int K, v8f acc, int lane) {
    const int half = lane >> 4;   // 0: lanes 0-15, 1: lanes 16-31
    const int l    = lane & 15;
    const float* a_row = A + (size_t)l * lda;
#pragma unroll 8
    for (int kk = 0; kk < K; kk += 4) {
        v2f a, b;
        a.x = a_row[kk + 2 * half];
        a.y = a_row[kk + 2 * half + 1];
        b.x = B[(size_t)(kk + 2 * half) * ldb + l];
        b.y = B[(size_t)(kk + 2 * half + 1) * ldb + l];
        acc = __builtin_amdgcn_wmma_f32_16x16x4_f32(false, a, false, b,
                                                    (short)0, acc, false, false);
    }
    return acc;
}

// ---------------- init: amax = enc(-inf), denom = 0, agg = 0 ----------------
__global__ void k_init(float* base, size_t nd) {
    size_t i = (size_t)blockIdx.x * blockDim.x + threadIdx.x;
    if (i < nd) {
        ((unsigned*)base)[i] = 0x007FFFFFu;   // fenc(-inf)
    } else if (i < 3 * nd) {
        base[i] = 0.0f;
    }
}

// ---------------- fused node GEMM: [u|v|k|h6] = h @ [W1|W2|W4|W6] ----------------
// grid(N/16), block(1024): 32 waves, wave w handles colTile w in [0,32); A tile via TDM.
__global__ void k_node_gemm4(const float* __restrict__ h,
                             const float* __restrict__ W1, const float* __restrict__ W2,
                             const float* __restrict__ W4, const float* __restrict__ W6,
                             float* __restrict__ u, float* __restrict__ v,
                             float* __restrict__ kv, float* __restrict__ h6) {
    __shared__ __align__(16) float s_A[16 * DD];
    const int mt   = blockIdx.x;
    const int tid  = threadIdx.x;
    const int wave = tid >> 5;
    const int lane = tid & 31;

    stage_rows(s_A, h + (size_t)mt * 16 * DD, 16, DD, DD, tid, 1024);

    const int colTile = wave;                           // 0..31
    const int wsel = colTile >> 3;                      // 0..3
    const int ncol = (colTile & 7) * 16;
    const float* W = (wsel == 0) ? W1 : (wsel == 1) ? W2 : (wsel == 2) ? W4 : W6;
    float* out     = (wsel == 0) ? u  : (wsel == 1) ? v  : (wsel == 2) ? kv : h6;

    v8f c = {};
    c = gemm16x16(s_A, DD, W + ncol, DD, DD, c, lane);

    const int half = lane >> 4, l = lane & 15;
#pragma unroll
    for (int r = 0; r < 8; r++) {
        int row = mt * 16 + r + 8 * half;
        out[(size_t)row * DD + ncol + l] = c[r];
    }
}

// ---------------- edge pass 1: attn logits + atomic segment max ----------------
// grid(E/16), block(256): wave w computes cols [16w,16w+16) of x3 for 16 edges.
__global__ void k_edge_pass1(const float* __restrict__ e_ij, const float* __restrict__ W3,
                             const float* __restrict__ u, const float* __restrict__ v,
                             const int* __restrict__ src, const int* __restrict__ dst,
                             unsigned* __restrict__ amax) {
    __shared__ __align__(16) float s_A[16 * DD];
    __shared__ int s_src[16], s_dst[16];
    const int et  = blockIdx.x;
    const int tid = threadIdx.x;
    if (tid < 16) { s_src[tid] = src[et * 16 + tid]; s_dst[tid] = dst[et * 16 + tid]; }
    stage_rows(s_A, e_ij + (size_t)et * 16 * DD, 16, DD, DD, tid, 256);

    const int wave = tid >> 5, lane = tid & 31;
    const int half = lane >> 4, l = lane & 15;
    const int col0 = wave * 16;

    if (et + 1 < gridDim.x)
        __builtin_prefetch(e_ij + (size_t)(et + 1) * 16 * DD + lane * 16, 0, 1);

    v8f c = {};
    c = gemm16x16(s_A, DD, W3 + col0, DD, DD, c, lane);

    const float scale = 0.1767766952966369f; // 1/sqrt(32)
#pragma unroll
    for (int r = 0; r < 8; r++) {
        int m  = r + 8 * half;
        int es = s_src[m], ed = s_dst[m];
        int col = col0 + l;
        float uv = u[(size_t)es * DD + col];
        float vv = v[(size_t)ed * DD + col];
        float attn = uv * (vv + c[r]) * scale;
        atomicMax(amax + (size_t)ed * DD + col, fenc(attn));
    }
}

// ---------------- edge pass 2: ex = exp(attn-amax); denom += ex; agg += ex*(k[dst]+x5) ----
__global__ void k_edge_pass2(const float* __restrict__ e_ij,
                             const float* __restrict__ W3, const float* __restrict__ W5,
                             const float* __restrict__ u, const float* __restrict__ v,
                             const float* __restrict__ kv,
                             const int* __restrict__ src, const int* __restrict__ dst,
                             const unsigned* __restrict__ amax,
                             float* __restrict__ denom, float* __restrict__ agg) {
    __shared__ __align__(16) float s_A[16 * DD];
    __shared__ int s_src[16], s_dst[16];
    const int et  = blockIdx.x;
    const int tid = threadIdx.x;
    if (tid < 16) { s_src[tid] = src[et * 16 + tid]; s_dst[tid] = dst[et * 16 + tid]; }
    stage_rows(s_A, e_ij + (size_t)et * 16 * DD, 16, DD, DD, tid, 256);

    const int wave = tid >> 5, lane = tid & 31;
    const int half = lane >> 4, l = lane & 15;
    const int col0 = wave * 16;

    if (et + 1 < gridDim.x)
        __builtin_prefetch(e_ij + (size_t)(et + 1) * 16 * DD + lane * 16, 0, 1);

    v8f c3 = {}, c5 = {};
    c3 = gemm16x16(s_A, DD, W3 + col0, DD, DD, c3, lane);
    c5 = gemm16x16(s_A, DD, W5 + col0, DD, DD, c5, lane);

    const float scale = 0.1767766952966369f;
#pragma unroll
    for (int r = 0; r < 8; r++) {
        int m  = r + 8 * half;
        int es = s_src[m], ed = s_dst[m];
        int col = col0 + l;
        size_t di = (size_t)ed * DD + col;
        float uv = u[(size_t)es * DD + col];
        float vv = v[di];
        float attn = uv * (vv + c3[r]) * scale;
        float mx = fdec(amax[di]);
        float ex = __expf(attn - mx);
        float m2 = kv[di] + c5[r];
        atomicAdd(denom + di, ex);
        atomicAdd(agg + di, ex * m2);
    }
}

// ---------------- LayerNorm helper (block = 128 threads, one row) ----------------
__device__ __forceinline__ float layernorm128(float x, const float* __restrict__ g,
                                              const float* __restrict__ b,
                                              float* red, int tid) {
    red[tid] = x;
    __syncthreads();
    for (int s = 64; s > 0; s >>= 1) { if (tid < s) red[tid] += red[tid + s]; __syncthreads(); }
    float mu = red[0] * (1.0f / DD);
    __syncthreads();
    float d = x - mu;
    red[tid] = d * d;
    __syncthreads();
    for (int s = 64; s > 0; s >>= 1) { if (tid < s) red[tid] += red[tid + s]; __syncthreads(); }
    float var = red[0] * (1.0f / DD);
    __syncthreads();
    return d * rsqrtf(var + 1e-5f) * g[tid] + b[tid];
}

// h1 = LN(h@W6 + agg/denom)
__global__ void k_h1_ln(const float* __restrict__ h6, const float* __restrict__ agg,
                        const float* __restrict__ denom,
                        const float* __restrict__ g, const float* __restrict__ b,
                        float* __restrict__ h1) {
    __shared__ float red[DD];
    const int row = blockIdx.x, tid = threadIdx.x;
    size_t idx = (size_t)row * DD + tid;
    float dn = denom[idx];
    float a = (dn > 0.0f) ? agg[idx] / dn : 0.0f;
    float x = h6[idx] + a;
    h1[idx] = layernorm128(x, g, b, red, tid);
}

// t = relu(h1 @ mlp_w1)   [N,256]; grid(N/16), block(512): 16 waves = 16 col tiles
__global__ void k_mlp1(const float* __restrict__ h1, const float* __restrict__ mw1,
                       float* __restrict__ t) {
    __shared__ __align__(16) float s_A[16 * DD];
    const int mt   = blockIdx.x;
    const int tid  = threadIdx.x;
    const int wave = tid >> 5;
    const int lane = tid & 31;

    stage_rows(s_A, h1 + (size_t)mt * 16 * DD, 16, DD, DD, tid, 512);

    const int ncol = wave * 16;
    v8f c = {};
    c = gemm16x16(s_A, DD, mw1 + ncol, 2 * DD, DD, c, lane);

    const int half = lane >> 4, l = lane & 15;
#pragma unroll
    for (int r = 0; r < 8; r++) {
        int row = mt * 16 + r + 8 * half;
        t[(size_t)row * (2 * DD) + ncol + l] = fmaxf(c[r], 0.0f);
    }
}

// mlp_out = t @ mlp_w2   [N,128], K=256; grid(N/16), block(256): 8 waves = 8 col tiles
__global__ void k_mlp2(const float* __restrict__ t, const float* __restrict__ mw2,
                       float* __restrict__ mlp_out) {
    __shared__ __align__(16) float s_A[16 * 2 * DD];
    const int mt   = blockIdx.x;
    const int tid  = threadIdx.x;
    const int wave = tid >> 5;
    const int lane = tid & 31;

    stage_rows(s_A, t + (size_t)mt * 16 * (2 * DD), 16, 2 * DD, 2 * DD, tid, 256);

    const int ncol = wave * 16;
    v8f c = {};
    c = gemm16x16(s_A, 2 * DD, mw2 + ncol, DD, 2 * DD, c, lane);

    const int half = lane >> 4, l = lane & 15;
#pragma unroll
    for (int r = 0; r < 8; r++) {
        int row = mt * 16 + r + 8 * half;
        mlp_out[(size_t)row * DD + ncol + l] = c[r];
    }
}

// h2 = LN(h1 + mlp_out)
__global__ void k_final_ln(const float* __restrict__ h1, const float* __restrict__ mlp_out,
                           const float* __restrict__ g, const float* __restrict__ b,
                           float* __restrict__ out) {
    __shared__ float red[DD];
    const int row = blockIdx.x, tid = threadIdx.x;
    size_t idx = (size_t)row * DD + tid;
    float x = h1[idx] + mlp_out[idx];
    out[idx] = layernorm128(x, g, b, red, tid);
}

extern "C" void kernel_launch(void* const* d_in, const int* in_sizes, int n_in,
                              void* d_out, int out_size, void* d_ws, size_t ws_size,
                              hipStream_t stream) {
    (void)in_sizes; (void)n_in; (void)out_size; (void)ws_size;

    const float* h    = (const float*)d_in[0];
    const float* e_ij = (const float*)d_in[1];
    const float* W1   = (const float*)d_in[2];
    const float* W2   = (const float*)d_in[3];
    const float* W3   = (const float*)d_in[4];
    const float* W4   = (const float*)d_in[5];
    const float* W5   = (const float*)d_in[6];
    const float* W6   = (const float*)d_in[7];
    const float* ln_g = (const float*)d_in[8];
    const float* ln_b = (const float*)d_in[9];
    const float* mw1  = (const float*)d_in[10];
    const float* mw2  = (const float*)d_in[11];
    const int*   src  = (const int*)d_in[12];
    const int*   dst  = (const int*)d_in[13];
    float* out = (float*)d_out;

    const size_t ND = (size_t)NN * DD;   // 12.8M floats
    float* ws      = (float*)d_ws;
    float* u       = ws + 0 * ND;
    float* v       = ws + 1 * ND;
    float* kv      = ws + 2 * ND;
    float* h6      = ws + 3 * ND;
    float* amax_f  = ws + 4 * ND;        // stored as monotone-encoded uint
    float* denom   = ws + 5 * ND;
    float* agg     = ws + 6 * ND;
    float* h1      = ws + 7 * ND;
    float* mlp_out = ws + 8 * ND;
    float* t       = ws + 9 * ND;        // N x 256 (2*ND floats)
    unsigned* amax = (unsigned*)amax_f;
    // total: 11*ND floats = 563.2 MB of workspace

    // 0) init amax/denom/agg (contiguous region starting at amax_f)
    {
        size_t n = 3 * ND;
        k_init<<<dim3((unsigned)((n + 255) / 256)), dim3(256), 0, stream>>>(amax_f, ND);
    }
    // 1) node GEMMs: u, v, k, h@W6 (one block = 32 waves = all 32 col tiles)
    k_node_gemm4<<<dim3(NN / 16), dim3(1024), 0, stream>>>(h, W1, W2, W4, W6, u, v, kv, h6);
    // 2) edge pass 1: segment max of logits
    k_edge_pass1<<<dim3(EE / 16), dim3(256), 0, stream>>>(e_ij, W3, u, v, src, dst, amax);
    // 3) edge pass 2: unnormalized softmax numerator/denominator
    k_edge_pass2<<<dim3(EE / 16), dim3(256), 0, stream>>>(e_ij, W3, W5, u, v, kv, src, dst,
                                                          amax, denom, agg);
    // 4) h1 = LN(h@W6 + agg/denom)
    k_h1_ln<<<dim3(NN), dim3(DD), 0, stream>>>(h6, agg, denom, ln_g, ln_b, h1);
    // 5) t = relu(h1 @ mlp_w1)
    k_mlp1<<<dim3(NN / 16), dim3(512), 0, stream>>>(h1, mw1, t);
    // 6) mlp_out = t @ mlp_w2
    k_mlp2<<<dim3(NN / 16), dim3(256), 0, stream>>>(t, mw2, mlp_out);
    // 7) h2 = LN(h1 + mlp_out)
    k_final_ln<<<dim3(NN), dim3(DD), 0, stream>>>(h1, mlp_out, ln_g, ln_b, out);
}